// AdditiveAttention_25469156065862
// MI455X (gfx1250) — compile-verified
//
#include <hip/hip_runtime.h>
#include <hip/hip_bf16.h>
#include <math.h>

// Problem constants (match reference)
#define BB  8
#define SQ  128
#define SK  1024
#define UU  256

typedef __attribute__((ext_vector_type(2))) float v2f;
typedef __attribute__((ext_vector_type(8))) float v8f;

#if __has_builtin(__builtin_amdgcn_tanhf)
  #define DEV_TANH(x) __builtin_amdgcn_tanhf(x)
#elif __has_builtin(__builtin_amdgcn_tanh_f32)
  #define DEV_TANH(x) __builtin_amdgcn_tanh_f32(x)
#else
  #define DEV_TANH(x) tanhf(x)
#endif

#if __has_builtin(__builtin_amdgcn_tensor_load_to_lds) && \
    __has_builtin(__builtin_amdgcn_s_wait_tensorcnt)
  #define HAVE_TDM 1
#else
  #define HAVE_TDM 0
#endif

// ---------------------------------------------------------------------------
// Batched f32 GEMM: C[b] = A[b] (MxK, row-major) * B[b] (KxN, row-major).
// One wave32 computes one 16x16 tile with V_WMMA_F32_16X16X4_F32.
// Tile assignment is wave-uniform so EXEC is all-ones at every WMMA.
// ---------------------------------------------------------------------------
__global__ void wmma_gemm_f32(const float* __restrict__ A,
                              const float* __restrict__ Bm,
                              float* __restrict__ C,
                              int M, int N, int K,
                              long long sA, long long sB, long long sC) {
  const int batch = blockIdx.y;
  const float* Ab = A  + (long long)batch * sA;
  const float* Bb = Bm + (long long)batch * sB;
  float*       Cb = C  + (long long)batch * sC;

  const int wave = threadIdx.x >> 5;
  const int lane = threadIdx.x & 31;
  const int tilesN = N >> 4;
  const int totTiles = (M >> 4) * tilesN;
  const int tile = blockIdx.x * (blockDim.x >> 5) + wave;
  if (tile >= totTiles) return;  // uniform per-wave: whole wave exits together

  const int tm = (tile / tilesN) << 4;
  const int tn = (tile % tilesN) << 4;
  const int half = lane >> 4;    // lanes 0-15 hold K=0..1, lanes 16-31 K=2..3
  const int l15  = lane & 15;

  v8f acc = {};
  const float* arow = Ab + (long long)(tm + l15) * K;   // A: row per lane
  for (int k0 = 0; k0 < K; k0 += 4) {
    const int kb = k0 + (half << 1);
    v2f a, b;
    a.x = arow[kb];
    a.y = arow[kb + 1];
    b.x = Bb[(long long)kb       * N + tn + l15];       // B: col per lane
    b.y = Bb[(long long)(kb + 1) * N + tn + l15];
    acc = __builtin_amdgcn_wmma_f32_16x16x4_f32(
        /*neg_a=*/false, a, /*neg_b=*/false, b,
        /*c_mod=*/(short)0, acc, /*reuse_a=*/false, /*reuse_b=*/false);
  }

#pragma unroll
  for (int j = 0; j < 8; ++j) {
    const int m = tm + j + (half << 3);                 // VGPR j: M=j / M=j+8
    Cb[(long long)m * N + tn + l15] = acc[j];
  }
}

#if HAVE_TDM
typedef __attribute__((ext_vector_type(4))) unsigned int u32x4;
typedef __attribute__((ext_vector_type(8))) int          i32x8;
typedef __attribute__((ext_vector_type(4))) int          i32x4;

// Issue one TDM DMA: copy 8 KB (2048 f32, contiguous) from global to LDS.
// D# built per CDNA5 ISA 8.3/8.4: 1D tile, data_size=4B, tile_dim0=2048.
// This toolchain's builtin is the 6-arg form:
//   (uint32x4 g0, int32x8 g1, int32x4 g2, int32x4 g3, int32x8 g4, i32 cpol)
__device__ __forceinline__ void tdm_copy_8kb(const float* __restrict__ src,
                                             void* lds_dst) {
  const unsigned long long ga = (unsigned long long)(uintptr_t)src;
  const unsigned lds_off = (unsigned)(uintptr_t)lds_dst;  // low 32 bits = LDS byte offset

  u32x4 g0;
  g0[0] = 1u;                                      // count=1 (valid user D#)
  g0[1] = lds_off;                                 // lds_addr
  g0[2] = (unsigned)(ga & 0xFFFFFFFFu);            // global_addr[31:0]
  g0[3] = (unsigned)((ga >> 32) & 0x01FFFFFFu)     // global_addr[56:32]
        | (2u << 30);                              // type = 2 ("image")

  const unsigned td0 = 2048u;                      // elements in dim0
  i32x8 g1;
  g1[0] = (int)(2u << 16);                         // wg_mask=0, data_size=2 (4B)
  g1[1] = (int)((td0 & 0xFFFFu) << 16);            // tensor_dim0[15:0] @ bits 63:48
  g1[2] = (int)(((td0 >> 16) & 0xFFFFu)            // tensor_dim0[31:16]
              | (1u << 16));                       // tensor_dim1 = 1
  g1[3] = (int)((td0 & 0xFFFFu) << 16);            // tile_dim0 = 2048 @ bits 127:112
  g1[4] = 0;                                       // tile_dim1=0 (unused), tile_dim2=0
  g1[5] = (int)td0;                                // tensor_dim0_stride (low 32)
  g1[6] = 0;
  g1[7] = 0;

  i32x4 z4 = {0, 0, 0, 0};                         // groups 2/3 unused (<=2D)
  i32x8 z8 = {0, 0, 0, 0, 0, 0, 0, 0};
  __builtin_amdgcn_tensor_load_to_lds(g0, g1, z4, z4, z8, 0);
}
#endif  // HAVE_TDM

// ---------------------------------------------------------------------------
// Fused additive-attention scores + softmax.
// One block (256 threads = 8 waves) per (b, q) row.
//   score[k] = sum_u wv[u] * tanh(qp[b,q,u] + kp[b,k,u])
//   attn[k]  = softmax_k(score)
// TDM path: wave 0 DMAs the next 8 k-rows (8 KB) into an LDS double buffer
// while all 8 waves compute tanh on the current buffer; s_wait_tensorcnt +
// barrier flips buffers. Fallback path streams k-rows with vector loads.
// ---------------------------------------------------------------------------
__global__ __launch_bounds__(256)
void additive_score_softmax(const float* __restrict__ qp,
                            const float* __restrict__ kp,
                            const float* __restrict__ wv,
                            float* __restrict__ attn) {
  __shared__ float s_q[UU];
  __shared__ float s_wv[UU];
  __shared__ float s_scores[SK];
  __shared__ float s_red[8];
#if HAVE_TDM
  __shared__ float s_kbuf[2][8][UU];   // 16 KB double buffer of k-rows
#endif

  const int row  = blockIdx.x;          // row = b*SQ + q
  const int b    = row >> 7;            // SQ == 128
  const int tid  = threadIdx.x;
  const int wave = tid >> 5;
  const int lane = tid & 31;

  s_q[tid]  = qp[(long long)row * UU + tid];
  s_wv[tid] = wv[tid];

  const float* kbase = kp + (long long)b * SK * UU;
  const int nt = SK / 8;                // 128 tiles of 8 k-rows

  // ---- scores -------------------------------------------------------------
#if HAVE_TDM
  if (wave == 0) {
    tdm_copy_8kb(kbase, &s_kbuf[0][0][0]);
    __builtin_amdgcn_s_wait_tensorcnt(0);
  }
  __syncthreads();

  for (int kt = 0; kt < nt; ++kt) {
    const int cur = kt & 1;
    if (wave == 0 && (kt + 1) < nt)
      tdm_copy_8kb(kbase + (long long)(kt + 1) * 8 * UU, &s_kbuf[1 - cur][0][0]);

    const int k = (kt << 3) + wave;     // this wave's k-row in current buffer
    float partial = 0.f;
#pragma unroll
    for (int j = 0; j < UU / 32; ++j) {
      const int u = lane + (j << 5);
      partial = fmaf(s_wv[u], DEV_TANH(s_q[u] + s_kbuf[cur][wave][u]), partial);
    }
#pragma unroll
    for (int off = 16; off >= 1; off >>= 1)
      partial += __shfl_xor(partial, off, 32);
    if (lane == 0) s_scores[k] = partial;

    if (wave == 0 && (kt + 1) < nt)
      __builtin_amdgcn_s_wait_tensorcnt(0);
    __syncthreads();
  }
#else
  __syncthreads();
  for (int kt = 0; kt < nt; ++kt) {
    const int k = (kt << 3) + wave;
    const float* krow = kbase + (long long)k * UU;
    if (kt + 1 < nt)
      __builtin_prefetch(krow + 8 * UU + lane * 8, 0, 1);

    float partial = 0.f;
#pragma unroll
    for (int j = 0; j < UU / 32; ++j) {
      const int u = lane + (j << 5);
      partial = fmaf(s_wv[u], DEV_TANH(s_q[u] + krow[u]), partial);
    }
#pragma unroll
    for (int off = 16; off >= 1; off >>= 1)
      partial += __shfl_xor(partial, off, 32);
    if (lane == 0) s_scores[k] = partial;
  }
  __syncthreads();
#endif

  // ---- softmax over SK=1024 ----------------------------------------------
  float m = -INFINITY;
  for (int k = tid; k < SK; k += 256) m = fmaxf(m, s_scores[k]);
#pragma unroll
  for (int off = 16; off >= 1; off >>= 1)
    m = fmaxf(m, __shfl_xor(m, off, 32));
  if (lane == 0) s_red[wave] = m;
  __syncthreads();
  if (tid == 0) {
    float mm = s_red[0];
#pragma unroll
    for (int w = 1; w < 8; ++w) mm = fmaxf(mm, s_red[w]);
    s_red[0] = mm;
  }
  __syncthreads();
  m = s_red[0];

  float sum = 0.f;
  for (int k = tid; k < SK; k += 256) {
    const float e = __expf(s_scores[k] - m);
    s_scores[k] = e;
    sum += e;
  }
#pragma unroll
  for (int off = 16; off >= 1; off >>= 1)
    sum += __shfl_xor(sum, off, 32);
  if (lane == 0) s_red[wave] = sum;
  __syncthreads();
  if (tid == 0) {
    float t = 0.f;
#pragma unroll
    for (int w = 0; w < 8; ++w) t += s_red[w];
    s_red[0] = 1.0f / t;
  }
  __syncthreads();
  const float inv = s_red[0];

  float* arow = attn + (long long)row * SK;
  for (int k = tid; k < SK; k += 256) arow[k] = s_scores[k] * inv;
}

// ---------------------------------------------------------------------------
extern "C" void kernel_launch(void* const* d_in, const int* in_sizes, int n_in,
                              void* d_out, int out_size, void* d_ws, size_t ws_size,
                              hipStream_t stream) {
  (void)in_sizes; (void)n_in; (void)out_size; (void)ws_size;

  const float* query = (const float*)d_in[0];  // [B,SQ,U]
  const float* key   = (const float*)d_in[1];  // [B,SK,U]
  const float* value = (const float*)d_in[2];  // [B,SK,U]
  const float* Wq    = (const float*)d_in[3];  // [U,U]
  const float* Wk    = (const float*)d_in[4];  // [U,U]
  const float* wv    = (const float*)d_in[5];  // [U]

  float* ctx  = (float*)d_out;                         // [B,SQ,U]
  float* attn = ctx + (size_t)BB * SQ * UU;            // [B,SQ,SK]

  float* qp = (float*)d_ws;                            // [B*SQ, U]  (1 MB)
  float* kp = qp + (size_t)BB * SQ * UU;               // [B*SK, U]  (8 MB)

  // 1) q-projection: [1024,256] = [1024,256] x [256,256]
  {
    const int M = BB * SQ, N = UU, K = UU;
    const int tiles = (M / 16) * (N / 16);             // 1024
    wmma_gemm_f32<<<dim3(tiles / 4, 1), 128, 0, stream>>>(
        query, Wq, qp, M, N, K, 0, 0, 0);
  }
  // 2) k-projection: [8192,256] = [8192,256] x [256,256]
  {
    const int M = BB * SK, N = UU, K = UU;
    const int tiles = (M / 16) * (N / 16);             // 8192
    wmma_gemm_f32<<<dim3(tiles / 4, 1), 128, 0, stream>>>(
        key, Wk, kp, M, N, K, 0, 0, 0);
  }
  // 3) fused tanh-score + softmax -> attention weights (TDM double-buffered)
  additive_score_softmax<<<dim3(BB * SQ), 256, 0, stream>>>(qp, kp, wv, attn);

  // 4) context: per batch [128,256] = [128,1024] x [1024,256]
  {
    const int M = SQ, N = UU, K = SK;
    const int tiles = (M / 16) * (N / 16);             // 128
    wmma_gemm_f32<<<dim3(tiles / 4, BB), 128, 0, stream>>>(
        attn, value, ctx, M, N, K,
        (long long)SQ * SK, (long long)SK * UU, (long long)SQ * UU);
  }
}